// UserEncoder_MINM_84026740179499
// MI455X (gfx1250) — compile-verified
//
#include <hip/hip_runtime.h>
#include <math.h>

// ---- model constants (from reference) ----
#define NUM_HEADS 2
#define HEADD     25
#define ATT_DIM   50
#define CUU       128
#define MEMN      64
#define MVDN      64
#define TLN       20
#define SHARPF    2.0f
#define CLIPF     20.0f

#define BPW          16     // batch elements per scan workgroup (WMMA M dimension)
#define SCAN_THREADS 256
#define NWAVES       (SCAN_THREADS / 32)
#define ATT_THREADS  256
#define HISC         200    // compile-time HIS for static LDS in attention

typedef float v2f __attribute__((ext_vector_type(2)));
typedef float v8f __attribute__((ext_vector_type(8)));

#if defined(__AMDGCN__) && __has_builtin(__builtin_amdgcn_wmma_f32_16x16x4_f32)
#define USE_WMMA 1
#else
#define USE_WMMA 0
#endif

__device__ __forceinline__ float sigm(float x) { return 1.0f / (1.0f + expf(-x)); }

// =====================================================================
// D[16][ldd] (LDS) = A[16][lda] (LDS, zero-padded K) @ W^T (+bias, clip)
// W: global, row-major [Nw][Kw]  (i.e. out = vec @ W^T, as in the reference)
// Uses V_WMMA_F32_16X16X4_F32: A is 16x4 f32 (2 VGPRs), B is 4x16 (2 VGPRs),
// C/D 16x16 f32 (8 VGPRs). Layouts per CDNA5 ISA 7.12.2.
// The tile loop index is forced into an SGPR (readfirstlane) so the loop is
// scalar and EXEC is all-ones at every WMMA, as the ISA requires.
// =====================================================================
__device__ __forceinline__ void wg_gemm(const float* __restrict__ W, int Kw, int Nw,
                                        const float* A, int lda,
                                        float* D, int ldd,
                                        const float* __restrict__ bias, int doClip,
                                        int tid)
{
#if USE_WMMA
  const int wave  = __builtin_amdgcn_readfirstlane(tid >> 5);  // wave-uniform -> SGPR
  const int lane  = tid & 31;
  const int mrow  = lane & 15;       // A: M index (batch).  B/D: column offset in tile
  const int hi    = lane >> 4;       // 0: K{0,1}/rows 0..7   1: K{2,3}/rows 8..15
  const int khalf = hi << 1;
  const int nTiles = (Nw + 15) >> 4;
  const int kSteps = (Kw + 3) >> 2;
  for (int nt = wave; nt < nTiles; nt += NWAVES) {
    const int n0 = nt << 4;
    const int nc = n0 + mrow;        // this lane's output column
    const float* Wrow = W + (size_t)(nc < Nw ? nc : 0) * (size_t)Kw;
    if (nc < Nw) __builtin_prefetch(Wrow, 0, 1);
    v8f acc = {0.f, 0.f, 0.f, 0.f, 0.f, 0.f, 0.f, 0.f};
    for (int ks = 0; ks < kSteps; ++ks) {
      const int k0 = ks << 2;
      const int kb = k0 + khalf;
      v2f a, b;
      a.x = A[mrow * lda + kb];                 // A zero-padded beyond Kw
      a.y = A[mrow * lda + kb + 1];
      b.x = (nc < Nw && kb     < Kw) ? Wrow[kb]     : 0.0f;  // B[k][n] = W[n][k]
      b.y = (nc < Nw && kb + 1 < Kw) ? Wrow[kb + 1] : 0.0f;
      acc = __builtin_amdgcn_wmma_f32_16x16x4_f32(false, a, false, b,
                                                  (short)0, acc, false, false);
    }
    if (nc < Nw) {
      const float bv = bias ? bias[nc] : 0.0f;
      const int mbase = hi << 3;     // C/D: VGPR r holds rows r (lanes 0-15), r+8 (16-31)
#pragma unroll
      for (int r = 0; r < 8; ++r) {
        float v = acc[r] + bv;
        if (doClip) v = fminf(fmaxf(v, -CLIPF), CLIPF);
        D[(mbase + r) * ldd + nc] = v;
      }
    }
  }
#else
  for (int idx = tid; idx < BPW * Nw; idx += SCAN_THREADS) {
    const int bb = idx / Nw, n = idx - bb * Nw;
    const float* Wrow = W + (size_t)n * Kw;
    const float* Arow = A + bb * lda;
    float acc = bias ? bias[n] : 0.0f;
    for (int k = 0; k < Kw; ++k) acc += Arow[k] * Wrow[k];
    if (doClip) acc = fminf(fmaxf(acc, -CLIPF), CLIPF);
    D[bb * ldd + n] = acc;
  }
#endif
}

// =====================================================================
// Kernel 1: click[b,t,:] = mean_j emb[user_click[b,t,j], :]
// Only needed for batches that can reach the slow scan path (seq_len >= HIS).
// =====================================================================
__global__ void embed_kernel(const int* __restrict__ uc, const int* __restrict__ seq_len,
                             const float* __restrict__ emb, float* __restrict__ click,
                             int B, int HIS)
{
  const int row = blockIdx.x;              // b*HIS + t
  const int b = row / HIS;
  if (seq_len[b] < HIS) return;            // MF[b] == tanh(init_M): x irrelevant
  const int d = threadIdx.x;
  if (d >= ATT_DIM) return;
  const int* idx = uc + (size_t)row * TLN;
  float acc = 0.0f;
  for (int j = 0; j < TLN; ++j) acc += emb[(size_t)idx[j] * ATT_DIM + d];
  click[(size_t)row * ATT_DIM + d] = acc * (1.0f / (float)TLN);
}

// =====================================================================
// Kernel 2: per (batch, head) attention. S=200, HEAD=25, f32 VALU (2 GFLOP total).
// =====================================================================
struct AttSmem {
  float x[HISC][ATT_DIM];
  float kh[HISC][HEADD];
  float vh[HISC][HEADD];
};

__global__ __launch_bounds__(ATT_THREADS)
void attention_kernel(const float* __restrict__ click, const int* __restrict__ seq_len,
                      const float* __restrict__ wq, const float* __restrict__ wk,
                      const float* __restrict__ wv, float* __restrict__ att,
                      int B, int HIS)
{
  const int b = blockIdx.x >> 1;
  const int head = blockIdx.x & 1;
  const int sl = seq_len[b];
  if (sl < HIS) return;
  __shared__ AttSmem sm;
  const int tid = threadIdx.x;
  for (int idx = tid; idx < HIS * ATT_DIM; idx += ATT_THREADS)
    sm.x[idx / ATT_DIM][idx % ATT_DIM] = click[(size_t)b * HIS * ATT_DIM + idx];
  __syncthreads();
  for (int idx = tid; idx < HIS * HEADD; idx += ATT_THREADS) {
    const int s = idx / HEADD, d = idx % HEADD;
    const float* wkr = wk + (size_t)(head * HEADD + d) * ATT_DIM;
    const float* wvr = wv + (size_t)(head * HEADD + d) * ATT_DIM;
    float ak = 0.f, av = 0.f;
    for (int c = 0; c < ATT_DIM; ++c) { const float xv = sm.x[s][c]; ak += xv * wkr[c]; av += xv * wvr[c]; }
    sm.kh[s][d] = ak; sm.vh[s][d] = av;
  }
  __syncthreads();
  for (int s = tid; s < HIS; s += ATT_THREADS) {
    float q[HEADD];
    for (int d = 0; d < HEADD; ++d) {
      const float* wqr = wq + (size_t)(head * HEADD + d) * ATT_DIM;
      float a = 0.f;
      for (int c = 0; c < ATT_DIM; ++c) a += sm.x[s][c] * wqr[c];
      q[d] = a;
    }
    float mx = -3.4e38f;
    for (int t = 0; t < HIS; ++t) {
      float l = 0.f;
      for (int d = 0; d < HEADD; ++d) l += q[d] * sm.kh[t][d];
      l = l * 0.2f + ((t < sl) ? 1.0f : -1.0e9f);     // mask1: +1 valid, -1e9 invalid
      mx = fmaxf(mx, l);
    }
    float o[HEADD];
    for (int d = 0; d < HEADD; ++d) o[d] = 0.f;
    float ssum = 0.f;
    for (int t = 0; t < HIS; ++t) {
      float l = 0.f;
      for (int d = 0; d < HEADD; ++d) l += q[d] * sm.kh[t][d];
      l = l * 0.2f + ((t < sl) ? 1.0f : -1.0e9f);
      const float e = expf(l - mx);
      ssum += e;
      for (int d = 0; d < HEADD; ++d) o[d] += e * sm.vh[t][d];
    }
    const float inv = 1.0f / ssum;
    float* orow = att + ((size_t)b * HIS + s) * ATT_DIM + head * HEADD;
    for (int d = 0; d < HEADD; ++d) orow[d] = o[d] * inv;
  }
}

// =====================================================================
// Kernel 3: the 200-step scan. One WG = 16 batch elements (WMMA M=16).
// M state lives in d_out (updated in place). sagg/wagg are dead -> skipped.
// Fast path: if no batch in the WG has seq_len >= HIS, MF == tanh(init_M).
// =====================================================================
struct ScanSmem {
  float kM[MEMN][MVDN];      // tanh(init_key_M)
  float M0[MEMN][MVDN];      // tanh(init_M), used for masked resets
  float Mn[MEMN];            // row norms of kM
  float h [BPW][CUU];        // masked hidden state
  float hn[BPW][CUU];        // unmasked h_new (feeds o2p)
  float rv[BPW][MVDN];       // read vector
  float ci[BPW][116];        // [x(50) | rv(64) | pad(2)]  K padded to 4
  int   slen[BPW];
  union {
    struct { float gi[BPW][384]; float gh[BPW][384]; } g;
    struct {
      float params[BPW][258];
      float karr[2 * BPW][MVDN];
      float wts [2 * BPW][MVDN];
      float ev[BPW][MVDN];
      float av[BPW][MVDN];
      float kn[2 * BPW];
      float beta[2 * BPW];
    } p;
  } u;
};

__global__ __launch_bounds__(SCAN_THREADS)
void scan_kernel(const float* __restrict__ att, const int* __restrict__ seq_len,
                 const float* __restrict__ W_ih, const float* __restrict__ W_hh,
                 const float* __restrict__ b_ih, const float* __restrict__ b_hh,
                 const float* __restrict__ o2pW, const float* __restrict__ o2pb,
                 const float* __restrict__ initM, const float* __restrict__ keyM,
                 const float* __restrict__ initR,
                 float* __restrict__ out, int B, int HIS)
{
  __shared__ ScanSmem sm;
  const int tid = threadIdx.x;
  const int b0 = blockIdx.x * BPW;

  // Initialize M (== d_out) to M0 = tanh(init_M) for our batches. This is also
  // the final answer for every batch with seq_len < HIS (mask pins M to M0).
  for (int idx = tid; idx < BPW * MEMN * MVDN; idx += SCAN_THREADS)
    out[(size_t)b0 * MEMN * MVDN + idx] = tanhf(initM[idx & (MEMN * MVDN - 1)]);

  bool anyFull = false;                       // uniform across the block
  for (int i = 0; i < BPW; ++i)
    if (seq_len[b0 + i] >= HIS) anyFull = true;
  if (tid < BPW) sm.slen[tid] = seq_len[b0 + tid];
  if (!anyFull) return;                        // MF = M0 for the whole WG

  // ---- prologue: constants + state into LDS ----
  for (int idx = tid; idx < MEMN * MVDN; idx += SCAN_THREADS) {
    sm.kM[idx >> 6][idx & 63] = tanhf(keyM[idx]);
    sm.M0[idx >> 6][idx & 63] = tanhf(initM[idx]);
  }
  for (int idx = tid; idx < BPW * CUU; idx += SCAN_THREADS)
    sm.h[idx >> 7][idx & 127] = 0.0f;
  for (int idx = tid; idx < BPW * MVDN; idx += SCAN_THREADS)
    sm.rv[idx >> 6][idx & 63] = tanhf(initR[idx & 63]);
  for (int idx = tid; idx < BPW * 116; idx += SCAN_THREADS)
    sm.ci[idx / 116][idx % 116] = 0.0f;
  __syncthreads();
  if (tid < MEMN) {
    float s = 0.f;
    for (int d = 0; d < MVDN; ++d) { const float v = sm.kM[tid][d]; s += v * v; }
    sm.Mn[tid] = sqrtf(s);
  }
  __syncthreads();

  for (int t = 0; t < HIS; ++t) {
    // ci = [x_t | rv]
    for (int idx = tid; idx < BPW * ATT_DIM; idx += SCAN_THREADS) {
      const int bb = idx / ATT_DIM, c = idx % ATT_DIM;
      sm.ci[bb][c] = att[((size_t)(b0 + bb) * HIS + t) * ATT_DIM + c];
    }
    for (int idx = tid; idx < BPW * MVDN; idx += SCAN_THREADS)
      sm.ci[idx >> 6][ATT_DIM + (idx & 63)] = sm.rv[idx >> 6][idx & 63];
    __syncthreads();

    // GRU GEMMs on WMMA: gi = ci @ W_ih^T + b_ih ;  gh = h @ W_hh^T + b_hh
    wg_gemm(W_ih, ATT_DIM + MVDN, 3 * CUU, &sm.ci[0][0], 116, &sm.u.g.gi[0][0], 3 * CUU, b_ih, 0, tid);
    wg_gemm(W_hh, CUU,            3 * CUU, &sm.h [0][0], CUU, &sm.u.g.gh[0][0], 3 * CUU, b_hh, 0, tid);
    __syncthreads();

    // GRU activation; params use UNMASKED h_new, carry uses masked h_new
    for (int idx = tid; idx < BPW * CUU; idx += SCAN_THREADS) {
      const int bb = idx >> 7, j = idx & 127;
      const float r  = sigm(sm.u.g.gi[bb][j]       + sm.u.g.gh[bb][j]);
      const float z  = sigm(sm.u.g.gi[bb][128 + j] + sm.u.g.gh[bb][128 + j]);
      const float n  = tanhf(sm.u.g.gi[bb][256 + j] + r * sm.u.g.gh[bb][256 + j]);
      const float hnew = (1.0f - z) * n + z * sm.h[bb][j];
      sm.hn[bb][j] = hnew;
      sm.h[bb][j]  = (t < sm.slen[bb]) ? hnew : 0.0f;
    }
    __syncthreads();

    // params = clip(h_new @ o2p_W^T + o2p_b, +-20)   (overwrites gi/gh region)
    wg_gemm(o2pW, CUU, 2 * (MVDN + 1) + 2 * MVDN, &sm.hn[0][0], CUU,
            &sm.u.p.params[0][0], 258, o2pb, 1, tid);
    __syncthreads();

    // head keys (pair = bb*2 + {read,write}), erase/add vectors
    for (int idx = tid; idx < 2 * BPW * MVDN; idx += SCAN_THREADS) {
      const int pair = idx >> 6, d = idx & 63;
      const int bb = pair >> 1, off = (pair & 1) ? 65 : 0;
      sm.u.p.karr[pair][d] = tanhf(sm.u.p.params[bb][off + d]);
    }
    for (int idx = tid; idx < BPW * MVDN; idx += SCAN_THREADS) {
      const int bb = idx >> 6, d = idx & 63;
      sm.u.p.ev[bb][d] = sigm (sm.u.p.params[bb][130 + d]);
      sm.u.p.av[bb][d] = tanhf(sm.u.p.params[bb][194 + d]);
    }
    __syncthreads();
    if (tid < 2 * BPW) {
      const int pair = tid, bb = pair >> 1;
      float s = 0.f;
      for (int d = 0; d < MVDN; ++d) { const float v = sm.u.p.karr[pair][d]; s += v * v; }
      sm.u.p.kn[pair] = sqrtf(s);
      const float pb = sm.u.p.params[bb][(pair & 1) ? 129 : 64];
      sm.u.p.beta[pair] = (log1pf(expf(pb)) + 1.0f) * SHARPF;
    }
    __syncthreads();
    // cosine scores * beta
    for (int idx = tid; idx < 2 * BPW * MEMN; idx += SCAN_THREADS) {
      const int pair = idx >> 6, m = idx & 63;
      float dot = 0.f;
      for (int d = 0; d < MVDN; ++d) dot += sm.kM[m][d] * sm.u.p.karr[pair][d];
      sm.u.p.wts[pair][m] = sm.u.p.beta[pair] * dot / (sm.Mn[m] * sm.u.p.kn[pair] + 1e-8f);
    }
    __syncthreads();
    // softmax over memory slots
    if (tid < 2 * BPW) {
      float mx = -3.4e38f;
      for (int m = 0; m < MEMN; ++m) mx = fmaxf(mx, sm.u.p.wts[tid][m]);
      float s = 0.f;
      for (int m = 0; m < MEMN; ++m) { const float e = expf(sm.u.p.wts[tid][m] - mx); sm.u.p.wts[tid][m] = e; s += e; }
      const float inv = 1.0f / s;
      for (int m = 0; m < MEMN; ++m) sm.u.p.wts[tid][m] *= inv;
    }
    __syncthreads();
    // rv = w_r . M (pre-update M), then M update + mask reset, in one pass
    for (int col = tid; col < BPW * MVDN; col += SCAN_THREADS) {
      const int bb = col >> 6, d = col & 63;
      const int msk = (t < sm.slen[bb]);
      const float evv = sm.u.p.ev[bb][d], avv = sm.u.p.av[bb][d];
      const size_t base = (size_t)(b0 + bb) * MEMN * MVDN + d;
      float acc = 0.f;
      for (int m = 0; m < MEMN; ++m) {
        const float Mv = out[base + (size_t)m * MVDN];
        acc += sm.u.p.wts[bb * 2 + 0][m] * Mv;
        const float ww = sm.u.p.wts[bb * 2 + 1][m];
        const float Mu = Mv * (1.0f - ww * evv) + ww * avv;
        out[base + (size_t)m * MVDN] = msk ? Mu : sm.M0[m][d];
      }
      sm.rv[bb][d] = acc;           // rv is NOT masked in the reference
    }
    __syncthreads();
  }
}

// =====================================================================
extern "C" void kernel_launch(void* const* d_in, const int* in_sizes, int n_in,
                              void* d_out, int out_size, void* d_ws, size_t ws_size,
                              hipStream_t stream)
{
  (void)n_in; (void)out_size; (void)ws_size;
  const int*   uc    = (const int*)  d_in[0];
  const int*   sl    = (const int*)  d_in[1];
  const float* emb   = (const float*)d_in[2];
  const float* wq    = (const float*)d_in[3];
  const float* wk    = (const float*)d_in[4];
  const float* wv    = (const float*)d_in[5];
  const float* W_ih  = (const float*)d_in[6];
  const float* W_hh  = (const float*)d_in[7];
  const float* b_ih  = (const float*)d_in[8];
  const float* b_hh  = (const float*)d_in[9];
  const float* o2pW  = (const float*)d_in[10];
  const float* o2pb  = (const float*)d_in[11];
  const float* initM = (const float*)d_in[12];
  const float* keyM  = (const float*)d_in[13];
  const float* initR = (const float*)d_in[14];

  const int B   = in_sizes[1];
  const int HIS = in_sizes[0] / (B * TLN);

  float* click = (float*)d_ws;                               // [B][HIS][50]
  float* att   = click + (size_t)B * HIS * ATT_DIM;          // [B][HIS][50]
  float* outf  = (float*)d_out;                              // [B][64][64]

  embed_kernel<<<B * HIS, 64, 0, stream>>>(uc, sl, emb, click, B, HIS);
  attention_kernel<<<B * NUM_HEADS, ATT_THREADS, 0, stream>>>(click, sl, wq, wk, wv, att, B, HIS);
  scan_kernel<<<(B + BPW - 1) / BPW, SCAN_THREADS, 0, stream>>>(
      att, sl, W_ih, W_hh, b_ih, b_hh, o2pW, o2pb, initM, keyM, initR, outf, B, HIS);
}